// Discriminator_69260642615905
// MI455X (gfx1250) — compile-verified
//
#include <hip/hip_runtime.h>
#include <hip/hip_bf16.h>

#define N_NODES   50000
#define F_IN      128
#define HID       256
#define N_GRAPHS  64
#define NEG_SLOPE 0.2f

typedef float v2f __attribute__((ext_vector_type(2)));
typedef float v8f __attribute__((ext_vector_type(8)));

// ---- workspace layout (float offsets) ----
#define OFF_DEG   0
#define OFF_DINV  (OFF_DEG + N_NODES)                 // 50000
#define OFF_H     100000                              // 16B aligned (400000 bytes)
#define OFF_AGG   (OFF_H + N_NODES * HID)             // 12,900,000
#define OFF_POOL  (OFF_AGG + N_NODES * HID)           // 25,700,000
#define OFF_CNT   (OFF_POOL + N_GRAPHS * HID)         // 25,716,384
// total ~25.72M floats ~= 103 MB

__device__ __forceinline__ float leaky(float v) {
    return v >= 0.0f ? v : NEG_SLOPE * v;
}

// deg = 1 (self loop), pool/cnt = 0  -- re-run every launch (ws is not re-poisoned)
__global__ void k_init(float* __restrict__ ws) {
    int i = blockIdx.x * blockDim.x + threadIdx.x;
    if (i < N_NODES)        ws[OFF_DEG  + i] = 1.0f;
    if (i < N_GRAPHS * HID) ws[OFF_POOL + i] = 0.0f;
    if (i < N_GRAPHS)       ws[OFF_CNT  + i] = 0.0f;
}

// deg[dst] += ew  (per edge)
__global__ void k_deg(const int* __restrict__ ei, const float* __restrict__ ew,
                      float* __restrict__ deg, int nE) {
    int e = blockIdx.x * blockDim.x + threadIdx.x;
    if (e < nE) atomicAdd(&deg[ei[nE + e]], ew[e]);
}

__global__ void k_dinv(float* __restrict__ ws) {
    int i = blockIdx.x * blockDim.x + threadIdx.x;
    if (i < N_NODES) ws[OFF_DINV + i] = rsqrtf(ws[OFF_DEG + i]);  // deg >= 1
}

// h = x @ W1 via V_WMMA_F32_16X16X4_F32; epilogue also writes agg = h * dinv^2
// (self-loop contribution of the sym-normalized scatter).
// Block = 256 threads = 8 waves; wave computes a 16x64 output strip.
// Out-of-range rows: A index is CLAMPED (not zeroed) -- garbage in A row m only
// corrupts D row m, and the epilogue never stores rows >= N_NODES. This keeps
// the inner loop free of exec-mask manipulation (pure clause loads + WMMA).
__global__ void __launch_bounds__(256) k_gemm(const float* __restrict__ x,
                                              const float* __restrict__ W1,
                                              const float* __restrict__ dinv,
                                              float* __restrict__ h,
                                              float* __restrict__ agg) {
    const int lane = threadIdx.x & 31;
    const int wave = threadIdx.x >> 5;
    const int kh   = lane >> 4;      // K-half selector (0/1)
    const int ln   = lane & 15;      // row/col within half
    const int m0   = blockIdx.x * 128 + wave * 16;
    const int n0   = blockIdx.y * 64;

    const int m  = m0 + ln;
    const int mc = (m < N_NODES) ? m : (N_NODES - 1);   // clamped A row

    const float* __restrict__ arow = x + (size_t)mc * F_IN + 2 * kh;

    v8f acc[4] = {};

    #pragma unroll 4
    for (int k = 0; k < F_IN; k += 4) {
        // A fragment: lane holds row mc, K = k + 2*kh .. k + 2*kh + 1
        const v2f a = *reinterpret_cast<const v2f*>(arow + k);
        #pragma unroll
        for (int j = 0; j < 4; ++j) {
            const int col = n0 + j * 16 + ln;
            v2f b;
            b.x = W1[(size_t)(k + 2 * kh)     * HID + col];
            b.y = W1[(size_t)(k + 2 * kh + 1) * HID + col];
            acc[j] = __builtin_amdgcn_wmma_f32_16x16x4_f32(
                false, a, false, b, (short)0, acc[j], false, false);
        }
    }

    // D layout: VGPR r -> row m0 + r + 8*kh, col n0 + j*16 + ln
    #pragma unroll
    for (int r = 0; r < 8; ++r) {
        const int row = m0 + r + 8 * kh;
        if (row < N_NODES) {
            const float d  = dinv[row];
            const float d2 = d * d;
            #pragma unroll
            for (int j = 0; j < 4; ++j) {
                const int   col = n0 + j * 16 + ln;
                const float v   = acc[j][r];
                h  [(size_t)row * HID + col] = v;
                agg[(size_t)row * HID + col] = v * d2;
            }
        }
    }
}

// agg[dst] += h[src] * (dinv[src]*ew*dinv[dst])  -- 4 edges per block, 64 lanes/edge, float4
__global__ void __launch_bounds__(256) k_scatter(const int* __restrict__ ei,
                                                 const float* __restrict__ ew,
                                                 const float* __restrict__ dinv,
                                                 const float* __restrict__ h,
                                                 float* __restrict__ agg, int nE) {
    const int e    = blockIdx.x * 4 + (threadIdx.x >> 6);
    const int lane = threadIdx.x & 63;
    if (e >= nE) return;
    const int   r    = ei[e];
    const int   c    = ei[nE + e];
    const float norm = dinv[r] * ew[e] * dinv[c];
    const float4 hv  = *reinterpret_cast<const float4*>(h + (size_t)r * HID + lane * 4);
    float* dst = agg + (size_t)c * HID + lane * 4;
    atomicAdd(dst + 0, hv.x * norm);
    atomicAdd(dst + 1, hv.y * norm);
    atomicAdd(dst + 2, hv.z * norm);
    atomicAdd(dst + 3, hv.w * norm);
}

// per node: v = leaky(agg + b1); pool[batch[n]] += v; cnt[batch[n]] += 1
__global__ void __launch_bounds__(256) k_pool(const float* __restrict__ agg,
                                              const float* __restrict__ b1,
                                              const int* __restrict__ batch,
                                              float* __restrict__ pool,
                                              float* __restrict__ cnt) {
    const int n = blockIdx.x;
    const int t = threadIdx.x;
    const int g = batch[n];
    const float v = leaky(agg[(size_t)n * HID + t] + b1[t]);
    atomicAdd(&pool[g * HID + t], v);
    if (t == 0) atomicAdd(&cnt[g], 1.0f);
}

// per graph: z = [pool/cnt, emb[label]]; z2 = leaky(z@W2 + b2); out = z2@W3 + b3
__global__ void __launch_bounds__(256) k_head(const float* __restrict__ pool,
                                              const float* __restrict__ cnt,
                                              const float* __restrict__ emb,
                                              const int* __restrict__ labels,
                                              const float* __restrict__ W2,
                                              const float* __restrict__ b2,
                                              const float* __restrict__ W3,
                                              const float* __restrict__ b3,
                                              float* __restrict__ out) {
    __shared__ float zin[HID + HID / 2];   // 384
    __shared__ float red[HID];
    const int g = blockIdx.x;
    const int t = threadIdx.x;

    const float inv = 1.0f / fmaxf(cnt[g], 1.0f);
    zin[t] = pool[g * HID + t] * inv;
    if (t < HID / 2) zin[HID + t] = emb[labels[g] * (HID / 2) + t];
    __syncthreads();

    float s = b2[t];
    #pragma unroll 8
    for (int j = 0; j < HID + HID / 2; ++j) s += zin[j] * W2[j * HID + t];
    s = leaky(s);
    red[t] = s * W3[t];
    __syncthreads();

    for (int st = 128; st > 0; st >>= 1) {
        if (t < st) red[t] += red[t + st];
        __syncthreads();
    }
    if (t == 0) out[g] = red[0] + b3[0];
}

extern "C" void kernel_launch(void* const* d_in, const int* in_sizes, int n_in,
                              void* d_out, int out_size, void* d_ws, size_t ws_size,
                              hipStream_t stream) {
    const float* x      = (const float*)d_in[0];
    const int*   ei     = (const int*)  d_in[1];   // [2, E] row-major
    const float* ew     = (const float*)d_in[2];
    const int*   batch  = (const int*)  d_in[3];
    const int*   labels = (const int*)  d_in[4];
    const float* W1     = (const float*)d_in[5];
    const float* b1     = (const float*)d_in[6];
    const float* emb    = (const float*)d_in[7];
    const float* W2     = (const float*)d_in[8];
    const float* b2     = (const float*)d_in[9];
    const float* W3     = (const float*)d_in[10];
    const float* b3     = (const float*)d_in[11];

    float* ws   = (float*)d_ws;
    float* out  = (float*)d_out;
    const int nE = in_sizes[1] / 2;

    float* dinv = ws + OFF_DINV;
    float* h    = ws + OFF_H;
    float* agg  = ws + OFF_AGG;
    float* pool = ws + OFF_POOL;
    float* cnt  = ws + OFF_CNT;
    (void)ws_size; (void)n_in; (void)out_size;

    k_init<<<(N_NODES + 255) / 256, 256, 0, stream>>>(ws);
    k_deg <<<(nE + 255) / 256,      256, 0, stream>>>(ei, ew, ws + OFF_DEG, nE);
    k_dinv<<<(N_NODES + 255) / 256, 256, 0, stream>>>(ws);

    dim3 ggrid((N_NODES + 127) / 128, HID / 64);
    k_gemm<<<ggrid, 256, 0, stream>>>(x, W1, dinv, h, agg);

    k_scatter<<<(nE + 3) / 4, 256, 0, stream>>>(ei, ew, dinv, h, agg, nE);
    k_pool   <<<N_NODES,      256, 0, stream>>>(agg, b1, batch, pool, cnt);
    k_head   <<<N_GRAPHS,     256, 0, stream>>>(pool, cnt, emb, labels, W2, b2, W3, b3, out);
}